// TemporalGNN_5239860101780
// MI455X (gfx1250) — compile-verified
//
#include <hip/hip_runtime.h>
#include <math.h>

// Problem constants from the reference
#define NNODES 50000
#define NFEAT  8
#define NPER   12        // T
#define NHID   64
#define FT     96        // NFEAT * NPER

typedef float v2f __attribute__((ext_vector_type(2)));
typedef float v8f __attribute__((ext_vector_type(8)));

__device__ __forceinline__ v8f wmma4(v2f a, v2f b, v8f c) {
  // D = A(16x4,f32) * B(4x16,f32) + C(16x16,f32)
  return __builtin_amdgcn_wmma_f32_16x16x4_f32(false, a, false, b, (short)0, c,
                                               false, false);
}

// ---------------- sparse phase ----------------

__global__ void fill_deg_kernel(float* deg, int n) {
  int i = blockIdx.x * blockDim.x + threadIdx.x;
  if (i < n) deg[i] = 1.0f;  // self-loop weight
}

__global__ void edge_deg_kernel(const int* __restrict__ ei,
                                const float* __restrict__ ew,
                                float* deg, int ne) {
  int i = blockIdx.x * blockDim.x + threadIdx.x;
  if (i < ne) atomicAdd(&deg[ei[ne + i]], ew[i]);  // dst row of edge_index
}

__global__ void dinv_kernel(float* deg, int n) {
  int i = blockIdx.x * blockDim.x + threadIdx.x;
  if (i < n) {
    float d = deg[i];
    deg[i] = (d > 0.0f) ? rsqrtf(d) : 0.0f;  // in place: deg -> dinv
  }
}

__global__ void init_agg_kernel(const float* __restrict__ x,
                                const float* __restrict__ dinv,
                                float* __restrict__ agg, int total) {
  int i = blockIdx.x * blockDim.x + threadIdx.x;
  if (i < total) {
    float di = dinv[i / FT];          // self-loop: dinv[n]*1*dinv[n]*x[n]
    agg[i] = di * di * x[i];
  }
}

__global__ void edge_agg_kernel(const int* __restrict__ ei,
                                const float* __restrict__ ew,
                                const float* __restrict__ dinv,
                                const float* __restrict__ x,
                                float* __restrict__ agg, int ne) {
  int i = blockIdx.x * blockDim.x + threadIdx.x;
  if (i >= ne) return;
  int s = ei[i];
  int d = ei[ne + i];
  float c = dinv[s] * ew[i] * dinv[d];
  const float4* xs = reinterpret_cast<const float4*>(x + (size_t)s * FT);
  float* ad = agg + (size_t)d * FT;
#pragma unroll 4
  for (int q = 0; q < FT / 4; ++q) {
    float4 v = xs[q];
    atomicAdd(ad + q * 4 + 0, c * v.x);
    atomicAdd(ad + q * 4 + 1, c * v.y);
    atomicAdd(ad + q * 4 + 2, c * v.z);
    atomicAdd(ad + q * 4 + 3, c * v.w);
  }
}

__global__ void softmax_kernel(const float* __restrict__ att,
                               float* __restrict__ probs) {
  if (threadIdx.x == 0 && blockIdx.x == 0) {
    float mx = att[0];
    for (int t = 1; t < NPER; ++t) mx = fmaxf(mx, att[t]);
    float e[NPER], s = 0.0f;
    for (int t = 0; t < NPER; ++t) { e[t] = __expf(att[t] - mx); s += e[t]; }
    float inv = 1.0f / s;
    for (int t = 0; t < NPER; ++t) probs[t] = e[t] * inv;
  }
}

// ---------------- dense WMMA phase ----------------
// One wave (32 lanes) per 16 nodes. N = 50000 = 3125 * 16 exactly.

__global__ __launch_bounds__(32) void dense_wmma_kernel(
    const float* __restrict__ agg, const float* __restrict__ convw,
    const float* __restrict__ convb, const float* __restrict__ linw,
    const float* __restrict__ linb, const float* __restrict__ probs,
    const float* __restrict__ clsw, const float* __restrict__ clsb,
    float* __restrict__ out) {
  __shared__ float s_convw[NFEAT * 192];  // (8 x 192)
  __shared__ float s_w0[NHID * NHID];     // lin_w[0][0:64,:]
  __shared__ float s_w2[NHID * NHID];     // lin_w[2][0:64,:]
  __shared__ float s_clsw[NHID * 16];     // (64 x 12) zero-padded to 16 cols
  __shared__ float s_convb[192];
  __shared__ float s_linb[128];           // [0:64]=lin_b[0], [64:128]=lin_b[2]
  __shared__ float s_agg[16 * FT];        // 16 nodes x (F,T)
  __shared__ float s_tmp[16 * 128];       // cz|ch spill (and Hacc reuse)

  const int lane = threadIdx.x;
  const int m0 = blockIdx.x * 16;

  for (int i = lane; i < NFEAT * 192; i += 32) s_convw[i] = convw[i];
  for (int i = lane; i < NHID * NHID; i += 32) {
    s_w0[i] = linw[i];                       // lin_w[0] rows 0..63
    s_w2[i] = linw[2 * 2 * NHID * NHID + i]; // lin_w[2] rows 0..63
  }
  for (int i = lane; i < NHID * 16; i += 32) {
    int r = i >> 4, c = i & 15;
    s_clsw[i] = (c < NPER) ? clsw[r * NPER + c] : 0.0f;
  }
  for (int i = lane; i < 192; i += 32) s_convb[i] = convb[i];
  for (int i = lane; i < 128; i += 32)
    s_linb[i] = (i < 64) ? linb[i] : linb[128 + (i - 64)];
  for (int i = lane; i < 16 * FT; i += 32) s_agg[i] = agg[(size_t)m0 * FT + i];
  __syncthreads();

  // WMMA f32 16x16x4 operand indexing (wave32)
  const int mrow = lane & 15;          // A-matrix row
  const int khi = (lane >> 4) * 2;     // A/B K sub-offset: 0 (lanes 0-15) / 2
  const int ncol = lane & 15;          // B/C/D column within tile
  const int mbase = (lane >> 4) * 8;   // C/D row offset (M = j + mbase)

  v8f hacc[4];
#pragma unroll
  for (int nt = 0; nt < 4; ++nt)
    hacc[nt] = (v8f){0.f, 0.f, 0.f, 0.f, 0.f, 0.f, 0.f, 0.f};

#pragma unroll 1
  for (int t = 0; t < NPER; ++t) {
    const float pt = probs[t];

    // Stage 1: cz = agg_t @ conv_w[:,0:64], ch = agg_t @ conv_w[:,128:192]
#pragma unroll
    for (int nt = 0; nt < 4; ++nt) {
      v8f az = (v8f){0.f, 0.f, 0.f, 0.f, 0.f, 0.f, 0.f, 0.f};
      v8f ah = az;
#pragma unroll
      for (int kt = 0; kt < 2; ++kt) {
        int k = 4 * kt + khi;
        v2f a, bz, bh;
        a.x = s_agg[mrow * FT + k * NPER + t];
        a.y = s_agg[mrow * FT + (k + 1) * NPER + t];
        bz.x = s_convw[k * 192 + nt * 16 + ncol];
        bz.y = s_convw[(k + 1) * 192 + nt * 16 + ncol];
        bh.x = s_convw[k * 192 + 128 + nt * 16 + ncol];
        bh.y = s_convw[(k + 1) * 192 + 128 + nt * 16 + ncol];
        az = wmma4(a, bz, az);
        ah = wmma4(a, bh, ah);
      }
      float cbz = s_convb[nt * 16 + ncol];
      float cbh = s_convb[128 + nt * 16 + ncol];
#pragma unroll
      for (int j = 0; j < 8; ++j) {  // spill to LDS to re-use as A operands
        s_tmp[(mbase + j) * 128 + nt * 16 + ncol] = az[j] + cbz;
        s_tmp[(mbase + j) * 128 + 64 + nt * 16 + ncol] = ah[j] + cbh;
      }
    }

    // Stage 2: Z = sigmoid(cz@W0+b0), Htilde = tanh(ch@W2+b2),
    //          Hacc += probs[t] * (1-Z) * Htilde  (R gate is dead: H==0)
#pragma unroll
    for (int nt = 0; nt < 4; ++nt) {
      v8f az = (v8f){0.f, 0.f, 0.f, 0.f, 0.f, 0.f, 0.f, 0.f};
      v8f ah = az;
#pragma unroll
      for (int kt = 0; kt < 16; ++kt) {
        int k = 4 * kt + khi;
        v2f acz, ach, b0, b2;
        acz.x = s_tmp[mrow * 128 + k];
        acz.y = s_tmp[mrow * 128 + k + 1];
        ach.x = s_tmp[mrow * 128 + 64 + k];
        ach.y = s_tmp[mrow * 128 + 64 + k + 1];
        b0.x = s_w0[k * 64 + nt * 16 + ncol];
        b0.y = s_w0[(k + 1) * 64 + nt * 16 + ncol];
        b2.x = s_w2[k * 64 + nt * 16 + ncol];
        b2.y = s_w2[(k + 1) * 64 + nt * 16 + ncol];
        az = wmma4(acz, b0, az);
        ah = wmma4(ach, b2, ah);
      }
      float b0v = s_linb[nt * 16 + ncol];
      float b2v = s_linb[64 + nt * 16 + ncol];
#pragma unroll
      for (int j = 0; j < 8; ++j) {
        float z = 1.0f / (1.0f + __expf(-(az[j] + b0v)));
        float ht = tanhf(ah[j] + b2v);
        hacc[nt][j] += pt * (1.0f - z) * ht;
      }
    }
  }

  // Classifier: out = relu(Hacc) @ cls_w + cls_b   (N-dim 12, padded to 16)
#pragma unroll
  for (int nt = 0; nt < 4; ++nt)
#pragma unroll
    for (int j = 0; j < 8; ++j)
      s_tmp[(mbase + j) * 64 + nt * 16 + ncol] = fmaxf(hacc[nt][j], 0.0f);

  v8f o = (v8f){0.f, 0.f, 0.f, 0.f, 0.f, 0.f, 0.f, 0.f};
#pragma unroll
  for (int kt = 0; kt < 16; ++kt) {
    int k = 4 * kt + khi;
    v2f a, b;
    a.x = s_tmp[mrow * 64 + k];
    a.y = s_tmp[mrow * 64 + k + 1];
    b.x = s_clsw[k * 16 + ncol];
    b.y = s_clsw[(k + 1) * 16 + ncol];
    o = wmma4(a, b, o);
  }
  if (ncol < NPER) {
    float cb = clsb[ncol];
#pragma unroll
    for (int j = 0; j < 8; ++j)
      out[(size_t)(m0 + mbase + j) * NPER + ncol] = o[j] + cb;
  }
}

// ---------------- launch ----------------

extern "C" void kernel_launch(void* const* d_in, const int* in_sizes, int n_in,
                              void* d_out, int out_size, void* d_ws,
                              size_t ws_size, hipStream_t stream) {
  const float* x = (const float*)d_in[0];
  const int* ei = (const int*)d_in[1];
  const float* ew = (const float*)d_in[2];
  const float* convw = (const float*)d_in[3];
  const float* convb = (const float*)d_in[4];
  const float* linw = (const float*)d_in[5];
  const float* linb = (const float*)d_in[6];
  const float* att = (const float*)d_in[7];
  const float* clsw = (const float*)d_in[8];
  const float* clsb = (const float*)d_in[9];
  float* out = (float*)d_out;

  const int N = in_sizes[0] / FT;   // 50000
  const int E = in_sizes[2];        // 1600000

  // workspace: deg/dinv (N) | agg (N*FT) | probs (NPER)   ~ 19.4 MB
  float* deg = (float*)d_ws;
  float* agg = deg + N;
  float* probs = agg + (size_t)N * FT;

  const int blk = 256;
  fill_deg_kernel<<<(N + blk - 1) / blk, blk, 0, stream>>>(deg, N);
  edge_deg_kernel<<<(E + blk - 1) / blk, blk, 0, stream>>>(ei, ew, deg, E);
  dinv_kernel<<<(N + blk - 1) / blk, blk, 0, stream>>>(deg, N);
  init_agg_kernel<<<((size_t)N * FT + blk - 1) / blk, blk, 0, stream>>>(
      x, deg, agg, N * FT);
  edge_agg_kernel<<<(E + blk - 1) / blk, blk, 0, stream>>>(ei, ew, deg, x, agg, E);
  softmax_kernel<<<1, 32, 0, stream>>>(att, probs);
  dense_wmma_kernel<<<N / 16, 32, 0, stream>>>(agg, convw, convb, linw, linb,
                                               probs, clsw, clsb, out);
}